// MeanResidueLossAdaptive_29643864277228
// MI455X (gfx1250) — compile-verified
//
#include <hip/hip_runtime.h>

typedef float v2f __attribute__((ext_vector_type(2)));
typedef float v8f __attribute__((ext_vector_type(8)));

#define W        101
#define EPS      1e-3f
#define RSTRIDE  106          // LDS row stride (floats): conflict-free for 16 rows, 8B aligned
#define NWAVES   8
#define RPW      16           // rows per wave (one 16x16 WMMA tile of rows)
#define RPB      (NWAVES * RPW)   // 128 rows per block

// D = A(16x4, f32) * B(4x16, f32) + C(16x16, f32), full f32 precision.
__device__ __forceinline__ v8f wmma4(v2f a, v2f b, v8f c) {
  // 8 args: (neg_a, A, neg_b, B, c_mod, C, reuse_a, reuse_b)
  return __builtin_amdgcn_wmma_f32_16x16x4_f32(false, a, false, b, (short)0, c,
                                               false, false);
}

__global__ __launch_bounds__(256) void mrl_stage1(const float* __restrict__ x,
                                                  const int* __restrict__ target,
                                                  float* __restrict__ part) {
  __shared__ __align__(16) float ebuf[NWAVES][RPW * RSTRIDE];  // exp(x) per row
  __shared__ float sumsA[NWAVES][32];   // per row: [sum_e, sum_e*a]
  __shared__ float sumsB[NWAVES][32];   // per row: [residue_sum, count_mask0]
  __shared__ float wacc[NWAVES][3];

  const int tid  = threadIdx.x;
  const int wave = tid >> 5;
  const int lane = tid & 31;
  const int half = lane >> 4;   // half-wave selects K pair (0,1) vs (2,3)
  const int n    = lane & 15;   // A row index / B column index

  const int rowBase = blockIdx.x * RPB + wave * RPW;
  const int row     = rowBase + n;                // row this lane owns
  const float* __restrict__ xrow = x + (size_t)row * W;
  const int tgt = target[row];

  float* erow = &ebuf[wave][n * RSTRIDE];

  // ---------------- Phase A: e = exp(x); WMMA-accumulate [sum_e | sum_e*k] ---
  // B col0 = ones (row sums), col1 = age weight k (weighted sums), else 0.
  float bw0 = (n == 0) ? 1.f : ((n == 1) ? (float)(2 * half) : 0.f);
  float bw1 = (n == 0) ? 1.f : ((n == 1) ? (float)(2 * half + 1) : 0.f);
  const float winc = (n == 1) ? 4.f : 0.f;

  v8f acc1 = {0.f, 0.f, 0.f, 0.f, 0.f, 0.f, 0.f, 0.f};

#pragma unroll 5
  for (int c = 0; c < 100; c += 4) {    // k = c + 2*half + {0,1} <= 99 < W: no guards
    const int k0 = c + 2 * half;
    const float x0 = xrow[k0];
    const float x1 = xrow[k0 + 1];
    const float e0 = __expf(x0);
    const float e1 = __expf(x1);
    erow[k0]     = e0;
    erow[k0 + 1] = e1;

    v2f A; A.x = e0;  A.y = e1;
    v2f B; B.x = bw0; B.y = bw1;
    acc1 = wmma4(A, B, acc1);
    bw0 += winc;
    bw1 += winc;
  }
  { // tail chunk c=100: only k=100 is a real bin (half==0, slot 0)
    const float x100 = xrow[100];               // in-bounds for every lane
    const float e0 = (half == 0) ? __expf(x100) : 0.f;
    const int k0 = 100 + 2 * half;
    erow[k0]     = e0;                          // zeros pad k=101..103
    erow[k0 + 1] = 0.f;

    v2f A; A.x = e0;  A.y = 0.f;                // zero A kills bogus weights
    v2f B; B.x = bw0; B.y = bw1;                // n==1: bw0 = 100 + 2*half
    acc1 = wmma4(A, B, acc1);
  }

  // e at the ground-truth bin: direct read of this row's staged values
  // (same-wave DS ops are in-order, so stores above are visible)
  const float egt = erow[tgt];

  // spill C columns 0,1 (lanes 0,1,16,17) so every lane can read its row's sums
  if (n < 2) {
#pragma unroll
    for (int vr = 0; vr < 8; ++vr)
      sumsA[wave][(vr + 8 * half) * 2 + n] = acc1[vr];
  }
  __syncthreads();

  const float sum_e = sumsA[wave][n * 2 + 0];
  const float sum_a = sumsA[wave][n * 2 + 1];
  const float inv   = 1.0f / sum_e;

  // ---------------- Phase B: residue terms + mask==0 count via WMMA ---------
  v2f Bc0; Bc0.x = (n == 0) ? 1.f : 0.f; Bc0.y = Bc0.x;  // col 0: residue
  v2f Bc1; Bc1.x = (n == 1) ? 1.f : 0.f; Bc1.y = Bc1.x;  // col 1: count
  v8f acc2 = {0.f, 0.f, 0.f, 0.f, 0.f, 0.f, 0.f, 0.f};

#pragma unroll 5
  for (int c = 0; c < 100; c += 4) {    // all k valid: no guards
    const int k0 = c + 2 * half;
    const float e0 = erow[k0];
    const float e1 = erow[k0 + 1];

    // mask = (p < p_gt) <=> (e < e_gt); p_not_K = mask * p
    const bool m0a = (e0 >= egt);
    const bool m1a = (e1 >= egt);
    const float q0 = (m0a ? 0.f : e0 * inv) + EPS;
    const float q1 = (m1a ? 0.f : e1 * inv) + EPS;
    const float t0 = -q0 * __logf(q0);
    const float t1 = -q1 * __logf(q1);
    const float g0 = m0a ? 1.f : 0.f;
    const float g1 = m1a ? 1.f : 0.f;

    v2f At; At.x = t0; At.y = t1;
    v2f Am; Am.x = g0; Am.y = g1;
    acc2 = wmma4(At, Bc0, acc2);  // col0 += residue terms
    acc2 = wmma4(Am, Bc1, acc2);  // col1 += (mask==0) count
  }
  { // tail: only k=100 contributes (half==0, slot 0)
    const float e0 = erow[100 + 2 * half];      // 0 for half==1
    const bool valid0 = (half == 0);
    const bool m0a = (e0 >= egt);
    const float q0 = (m0a ? 0.f : e0 * inv) + EPS;
    float t0 = -q0 * __logf(q0);
    t0 = valid0 ? t0 : 0.f;
    const float g0 = (valid0 && m0a) ? 1.f : 0.f;

    v2f At; At.x = t0; At.y = 0.f;
    v2f Am; Am.x = g0; Am.y = 0.f;
    acc2 = wmma4(At, Bc0, acc2);
    acc2 = wmma4(Am, Bc1, acc2);
  }

  if (n < 2) {
#pragma unroll
    for (int vr = 0; vr < 8; ++vr)
      sumsB[wave][(vr + 8 * half) * 2 + n] = acc2[vr];
  }
  __syncthreads();

  // ---------------- per-row finals + deterministic block reduction ----------
  float ml = 0.f, res = 0.f, cnt = 0.f;
  if (lane < 16) {
    res = sumsB[wave][lane * 2 + 0];
    cnt = sumsB[wave][lane * 2 + 1];
    const float mean = sum_a / sum_e;
    const float d = mean - (float)tgt;
    ml = d * d;
  }
#pragma unroll
  for (int off = 16; off >= 1; off >>= 1) {
    ml  += __shfl_xor(ml, off);
    res += __shfl_xor(res, off);
    cnt += __shfl_xor(cnt, off);
  }
  if (lane == 0) {
    wacc[wave][0] = ml; wacc[wave][1] = res; wacc[wave][2] = cnt;
  }
  __syncthreads();
  if (tid == 0) {
    float a = 0.f, b = 0.f, c = 0.f;
#pragma unroll
    for (int wv = 0; wv < NWAVES; ++wv) {
      a += wacc[wv][0]; b += wacc[wv][1]; c += wacc[wv][2];
    }
    part[blockIdx.x * 3 + 0] = a;
    part[blockIdx.x * 3 + 1] = b;
    part[blockIdx.x * 3 + 2] = c;
  }
}

__global__ __launch_bounds__(256) void mrl_stage2(const float* __restrict__ part,
                                                  int nblocks, float* __restrict__ out,
                                                  float invN) {
  __shared__ float red[256][3];
  float a = 0.f, b = 0.f, c = 0.f;
  for (int i = threadIdx.x; i < nblocks; i += 256) {
    a += part[i * 3 + 0];
    b += part[i * 3 + 1];
    c += part[i * 3 + 2];
  }
  red[threadIdx.x][0] = a; red[threadIdx.x][1] = b; red[threadIdx.x][2] = c;
  __syncthreads();
  for (int s = 128; s > 0; s >>= 1) {
    if (threadIdx.x < (unsigned)s) {
      red[threadIdx.x][0] += red[threadIdx.x + s][0];
      red[threadIdx.x][1] += red[threadIdx.x + s][1];
      red[threadIdx.x][2] += red[threadIdx.x + s][2];
    }
    __syncthreads();
  }
  if (threadIdx.x == 0) {
    out[0] = 0.2f * 0.5f * red[0][0] * invN;   // LAMBDA_1 * mean_loss
    out[1] = 0.05f * red[0][1] * invN;         // LAMBDA_2 * residue_loss
    out[2] = red[0][2] * invN;                 // batch_average_K
  }
}

extern "C" void kernel_launch(void* const* d_in, const int* in_sizes, int n_in,
                              void* d_out, int out_size, void* d_ws, size_t ws_size,
                              hipStream_t stream) {
  const float* x   = (const float*)d_in[0];
  const int*   tgt = (const int*)d_in[1];
  const int N = in_sizes[1];            // 524288 rows
  const int nblocks = N / RPB;          // 4096

  float* part = (float*)d_ws;           // nblocks*3 floats of scratch
  mrl_stage1<<<nblocks, 256, 0, stream>>>(x, tgt, part);
  mrl_stage2<<<1, 256, 0, stream>>>(part, nblocks, (float*)d_out,
                                    1.0f / (float)N);
}